// SinusoidalPositionalEmbedding_79491254714894
// MI455X (gfx1250) — compile-verified
//
#include <hip/hip_runtime.h>
#include <stdint.h>

typedef float v4f __attribute__((ext_vector_type(4)));
typedef int   v4i __attribute__((ext_vector_type(4)));

#define AS1 __attribute__((address_space(1)))
#define AS3 __attribute__((address_space(3)))

#if defined(__has_builtin)
#if __has_builtin(__builtin_amdgcn_global_load_async_to_lds_b128)
#define HAVE_ASYNC_BUILTIN 1
#endif
#if __has_builtin(__builtin_amdgcn_s_wait_asynccnt)
#define HAVE_WAIT_ASYNC_BUILTIN 1
#endif
#endif

// Async global -> LDS 16B copy (CDNA5 ASYNCcnt path).
__device__ __forceinline__ void async_copy_b128(const void* gsrc, void* ldst) {
#ifdef HAVE_ASYNC_BUILTIN
  __builtin_amdgcn_global_load_async_to_lds_b128(
      (AS1 v4i*)(uintptr_t)gsrc, (AS3 v4i*)ldst, 0, 0);
#else
  uint32_t loff  = (uint32_t)(uintptr_t)(AS3 void*)ldst;
  uint64_t gaddr = (uint64_t)(uintptr_t)gsrc;
  asm volatile("global_load_async_to_lds_b128 %0, %1, off"
               :: "v"(loff), "v"(gaddr)
               : "memory");
#endif
}

template <int N>
__device__ __forceinline__ void wait_asynccnt() {
#ifdef HAVE_WAIT_ASYNC_BUILTIN
  __builtin_amdgcn_s_wait_asynccnt(N);
#else
  asm volatile("s_wait_asynccnt %0" :: "i"(N) : "memory");
#endif
}

// ---------------------------------------------------------------------------
// Kernel C (placed first so the disasm snippet shows it): out = x + masked
// sinusoidal PE. Streaming float4 kernel with a double-buffered async
// global->LDS pipeline for x (each wave consumes only the LDS it staged
// itself, so only s_wait_asynccnt is needed, no barrier). Table values are
// recomputed on the fly: freq = exp(-m*scale), sin/cos(pos*freq).
// ---------------------------------------------------------------------------
__global__ __launch_bounds__(256)
void pe_add_kernel(const float* __restrict__ x,
                   const int* __restrict__ slen,
                   const int* __restrict__ sdim,
                   const int* __restrict__ lengths,
                   float* __restrict__ out,
                   unsigned total4) {
  __shared__ v4f stage[2][256];

  const int tid  = threadIdx.x;
  const int S    = *slen;
  const int D    = *sdim;
  const int half = D >> 1;
  const unsigned d4cnt = (unsigned)(D >> 2);
  const unsigned twoS  = 2u * (unsigned)S;
  // log(10000) / (half - 1)
  const float scale = 9.210340371976184f / (float)(half - 1);

  const unsigned stride = gridDim.x * blockDim.x;
  const unsigned base   = blockIdx.x * blockDim.x + tid;
  const unsigned nIter  = (total4 + stride - 1) / stride;
  const v4f* __restrict__ x4 = (const v4f*)x;
  v4f* __restrict__ out4 = (v4f*)out;

  if (nIter == 0) return;

  // Prologue: stage iteration 0 (index clamped so every lane always issues).
  {
    unsigned ci = base < total4 ? base : (total4 - 1u);
    async_copy_b128(&x4[ci], &stage[0][tid]);
  }

  for (unsigned k = 0; k < nIter; ++k) {
    const int cur = (int)(k & 1u);
    if (k + 1u < nIter) {
      unsigned nidx = base + (k + 1u) * stride;
      unsigned ci = nidx < total4 ? nidx : (total4 - 1u);
      async_copy_b128(&x4[ci], &stage[cur ^ 1][tid]);
      wait_asynccnt<1>();   // iteration k's stage is complete
    } else {
      wait_asynccnt<0>();
    }

    const v4f xv = stage[cur][tid];
    const unsigned i = base + k * stride;
    if (i < total4) {
      const unsigned row = i / d4cnt;
      const int d  = (int)(i - row * d4cnt) << 2;
      const unsigned b = row / twoS;
      const int s   = (int)(row - b * twoS);
      const int pos = (s >= S) ? (s - S) : s;

      v4f o = xv;
      if (pos < lengths[b]) {
        const float fpos = (float)pos;
#pragma unroll
        for (int j = 0; j < 4; ++j) {
          const int dd = d + j;
          const int m  = (dd < half) ? dd : (dd - half);
          const float freq = __expf(-scale * (float)m);
          const float ang  = fpos * freq;
          float v = (dd < half) ? __sinf(ang) : __cosf(ang);
          if (dd >= 2 * half) v = 0.0f;  // odd-D zero pad column
          o[j] += v;
        }
      }
      __builtin_nontemporal_store(o, &out4[i]);  // streaming store, spare L2
    }
  }
}

// ---------------------------------------------------------------------------
// Kernel A: zero per-sample length counters (ws is poisoned between runs).
// ---------------------------------------------------------------------------
__global__ void zero_lengths_kernel(int* __restrict__ lengths,
                                    const int* __restrict__ slen,
                                    const int* __restrict__ sdim,
                                    unsigned total) {
  const unsigned S = (unsigned)*slen;
  const unsigned D = (unsigned)*sdim;
  const unsigned B = total / (2u * S * D);
  for (unsigned i = threadIdx.x; i < B; i += blockDim.x) lengths[i] = 0;
}

// ---------------------------------------------------------------------------
// Kernel B: lengths[b] = #rows in x[b, :S, :] with any nonzero element.
// One wave32 per row; float4 loads; ballot any; per-wave accumulation keyed
// by sample index so atomics are ~1 per wave-chunk, not 1 per row.
// ---------------------------------------------------------------------------
__global__ __launch_bounds__(256)
void compute_lengths_kernel(const float* __restrict__ x,
                            const int* __restrict__ slen,
                            const int* __restrict__ sdim,
                            int* __restrict__ lengths,
                            unsigned total) {
  const int S = *slen;
  const int D = *sdim;
  const int B = (int)(total / (2u * (unsigned)S * (unsigned)D));
  const int R = B * S;              // rows to examine
  const int d4 = D >> 2;

  const int lane       = threadIdx.x & 31;
  const int wid        = (int)((blockIdx.x * blockDim.x + threadIdx.x) >> 5);
  const int wavesTotal = (int)((gridDim.x * blockDim.x) >> 5);
  const int rpw        = (R + wavesTotal - 1) / wavesTotal;
  int r0 = wid * rpw;
  int r1 = r0 + rpw;
  if (r1 > R) r1 = R;

  int curb = -1;
  int acc  = 0;
  for (int r = r0; r < r1; ++r) {
    const int b = r / S;
    const int p = r - b * S;
    const v4f* rp = (const v4f*)(x + ((size_t)b * 2u * (size_t)S + (size_t)p) * (size_t)D);
    bool nz = false;
    for (int c = lane; c < d4; c += 32) {
      v4f v = rp[c];
      nz |= (v.x != 0.0f) | (v.y != 0.0f) | (v.z != 0.0f) | (v.w != 0.0f);
    }
    const int anynz = __any((int)nz);
    if (lane == 0) {
      if (b != curb) {
        if (curb >= 0 && acc) atomicAdd(&lengths[curb], acc);
        curb = b;
        acc  = 0;
      }
      if (anynz) ++acc;
    }
  }
  if (lane == 0 && curb >= 0 && acc) atomicAdd(&lengths[curb], acc);
}

// ---------------------------------------------------------------------------
extern "C" void kernel_launch(void* const* d_in, const int* in_sizes, int n_in,
                              void* d_out, int out_size, void* d_ws, size_t ws_size,
                              hipStream_t stream) {
  const float* x    = (const float*)d_in[0];
  const int*   slen = (const int*)d_in[1];
  const int*   sdim = (const int*)d_in[2];
  float* out        = (float*)d_out;
  int* lengths      = (int*)d_ws;

  const unsigned total  = (unsigned)in_sizes[0];
  const unsigned total4 = total >> 2;
  if (total4 == 0) return;

  zero_lengths_kernel<<<1, 256, 0, stream>>>(lengths, slen, sdim, total);
  compute_lengths_kernel<<<1024, 256, 0, stream>>>(x, slen, sdim, lengths, total);

  // 8 float4 iterations per thread at full size (16.7M float4 -> 8192 blocks).
  const unsigned blocksC = (total4 + 2047u) / 2048u;
  pe_add_kernel<<<blocksC, 256, 0, stream>>>(x, slen, sdim, lengths, out, total4);
}